// VectorQuantizerEMA_9320079033229
// MI455X (gfx1250) — compile-verified
//
#include <hip/hip_runtime.h>
#include <hip/hip_bf16.h>

typedef float v2f __attribute__((ext_vector_type(2)));
typedef float v8f __attribute__((ext_vector_type(8)));

#define N_TOTAL   16384      // B*H*W = 16*32*32
#define C_DIM     256
#define V_DIM     4096
#define ZQ_ELEMS  4194304    // 16*256*32*32
#define PADK      258        // 256 + 2 pad floats: breaks LDS bank conflicts

// ---------------------------------------------------------------------------
// Kernel 1: per-codebook-row squared norms ||e_v||^2  (wave-per-row, wave32)
// ---------------------------------------------------------------------------
__global__ __launch_bounds__(256) void vq_enorm_kernel(const float* __restrict__ emb,
                                                       float* __restrict__ enorm) {
    const int lane = threadIdx.x & 31;
    const int row  = blockIdx.x * 8 + (threadIdx.x >> 5);
    const float* p = emb + (size_t)row * C_DIM;
    float s = 0.0f;
    #pragma unroll
    for (int k = lane; k < C_DIM; k += 32) {
        float v = p[k];
        s += v * v;
    }
    #pragma unroll
    for (int off = 16; off >= 1; off >>= 1)
        s += __shfl_xor(s, off, 32);
    if (lane == 0) enorm[row] = s;
}

// ---------------------------------------------------------------------------
// Kernel 2: fused distance GEMM (fp32 WMMA) + argmin over V
// Block = 256 threads = 8 waves; block handles 32 rows of h_flat.
// Each wave owns column group (wave) and computes BOTH 16-row tiles with two
// accumulators so every global B fragment feeds two v_wmma ops (2:1 ratio).
// score(m,v) = ||e_v||^2 - 2 * h_m . e_v    (||h||^2 dropped: row-constant)
// ---------------------------------------------------------------------------
__global__ __launch_bounds__(256) void vq_argmin_kernel(const float* __restrict__ h,
                                                        const float* __restrict__ emb,
                                                        const float* __restrict__ enorm,
                                                        int* __restrict__ out_idx) {
    __shared__ float sh[32 * PADK];       // 32 rows of h_flat, K=256 each (row-major, padded)
    __shared__ float red_s[32][8];
    __shared__ int   red_i[32][8];

    const int tid  = threadIdx.x;
    const int row0 = blockIdx.x * 32;     // first n-row handled by this block

    // Stage h_flat rows into LDS.  h is (B,C,H,W): h_flat[n][k] = h[b*262144 + k*1024 + (n%1024)]
    // i -> (m = i&31, k = i>>5): 32 consecutive tids cover 32 consecutive n at fixed k (coalesced).
    for (int i = tid; i < 32 * C_DIM; i += 256) {
        const int m  = i & 31;
        const int k  = i >> 5;
        const int n  = row0 + m;
        const int b  = n >> 10;
        const int hw = n & 1023;
        sh[m * PADK + k] = h[((size_t)b << 18) + ((size_t)k << 10) + (size_t)hw];
    }
    __syncthreads();

    const int lane  = tid & 31;
    const int cg    = tid >> 5;           // column group 0..7
    const int lmod  = lane & 15;
    const int lhalf = lane >> 4;

    // A fragment sources (ISA 16x4 f32 layout: lane M = lmod, K pair base = 2*lhalf)
    const float* aBase0 = &sh[(     lmod) * PADK + (lhalf << 1)];   // rows 0..15
    const float* aBase1 = &sh[(16 + lmod) * PADK + (lhalf << 1)];   // rows 16..31

    float best0[8], best1[8];
    int   bidx0[8], bidx1[8];
    #pragma unroll
    for (int r = 0; r < 8; ++r) {
        best0[r] = 3.402823466e+38f; bidx0[r] = 0;
        best1[r] = 3.402823466e+38f; bidx1[r] = 0;
    }

    for (int ct = cg; ct < 256; ct += 8) {
        const int col0  = ct << 4;
        const int mycol = col0 + lmod;            // lane's N column (B layout: lane = N)
        const float* bBase = emb + ((size_t)mycol << 8) + (lhalf << 1);

        v8f acc0 = {};
        v8f acc1 = {};
        #pragma unroll 16
        for (int kb = 0; kb < C_DIM; kb += 4) {
            v2f a0 = *(const v2f*)(aBase0 + kb);  // LDS, 8B aligned
            v2f a1 = *(const v2f*)(aBase1 + kb);  // LDS
            v2f b  = *(const v2f*)(bBase  + kb);  // global (L2-resident codebook), reused twice
            acc0 = __builtin_amdgcn_wmma_f32_16x16x4_f32(
                false, a0, false, b, (short)0, acc0, false, false);
            acc1 = __builtin_amdgcn_wmma_f32_16x16x4_f32(
                false, a1, false, b, (short)0, acc1, false, false);
        }

        const float en = enorm[mycol];
        #pragma unroll
        for (int r = 0; r < 8; ++r) {             // lane holds rows M = r + 8*lhalf, col mycol
            const float s0 = en - 2.0f * acc0[r];
            if (s0 < best0[r]) { best0[r] = s0; bidx0[r] = mycol; }
            const float s1 = en - 2.0f * acc1[r];
            if (s1 < best1[r]) { best1[r] = s1; bidx1[r] = mycol; }
        }
    }

    // Cross-lane argmin within each 16-lane column group (ties -> lowest index, like jnp.argmin)
    #pragma unroll
    for (int r = 0; r < 8; ++r) {
        #pragma unroll
        for (int off = 8; off >= 1; off >>= 1) {
            float os; int oi;
            os = __shfl_xor(best0[r], off, 16);
            oi = __shfl_xor(bidx0[r], off, 16);
            if (os < best0[r] || (os == best0[r] && oi < bidx0[r])) { best0[r] = os; bidx0[r] = oi; }
            os = __shfl_xor(best1[r], off, 16);
            oi = __shfl_xor(bidx1[r], off, 16);
            if (os < best1[r] || (os == best1[r] && oi < bidx1[r])) { best1[r] = os; bidx1[r] = oi; }
        }
    }

    // Lane lmod==0 of each half-wave holds rows (tile base) + 8*lhalf + r
    if (lmod == 0) {
        #pragma unroll
        for (int r = 0; r < 8; ++r) {
            const int lm = (lhalf << 3) + r;
            red_s[lm][cg]      = best0[r];
            red_i[lm][cg]      = bidx0[r];
            red_s[16 + lm][cg] = best1[r];
            red_i[16 + lm][cg] = bidx1[r];
        }
    }
    __syncthreads();

    // Final cross-wave (column-group) argmin; 32 rows -> first 32 threads
    if (tid < 32) {
        float bs = red_s[tid][0];
        int   bi = red_i[tid][0];
        #pragma unroll
        for (int c = 1; c < 8; ++c) {
            const float os = red_s[tid][c];
            const int   oi = red_i[tid][c];
            if (os < bs || (os == bs && oi < bi)) { bs = os; bi = oi; }
        }
        out_idx[row0 + tid] = bi;
    }
}

// ---------------------------------------------------------------------------
// Kernel 3: gather z_q (== z_q_st numerically), emit indices as float,
// and produce deterministic per-block MSE partial sums.
// 4096 blocks x 256 threads x 4 elems = 4,194,304
// ---------------------------------------------------------------------------
__global__ __launch_bounds__(256) void vq_gather_kernel(const float* __restrict__ h,
                                                        const float* __restrict__ emb,
                                                        const int* __restrict__ idx,
                                                        float* __restrict__ out_zq,
                                                        float* __restrict__ out_idx_f,
                                                        float* __restrict__ partials) {
    __shared__ float red[256];
    const size_t base = (size_t)blockIdx.x * 1024 + threadIdx.x;
    float local = 0.0f;
    #pragma unroll
    for (int i = 0; i < 4; ++i) {
        const size_t e  = base + (size_t)i * 256;
        const int b  = (int)(e >> 18);
        const int c  = (int)((e >> 10) & 255);
        const int hw = (int)(e & 1023);
        const int n  = (b << 10) + hw;
        const int q  = idx[n];
        const float zq = emb[((size_t)q << 8) + (size_t)c];
        out_zq[e] = zq;                 // z_q_st = h + (z_q - h) = z_q numerically
        const float d = h[e] - zq;
        local += d * d;
    }
    const int gid = blockIdx.x * 256 + threadIdx.x;
    if (gid < N_TOTAL) out_idx_f[gid] = (float)idx[gid];

    red[threadIdx.x] = local;
    __syncthreads();
    #pragma unroll
    for (int s = 128; s > 0; s >>= 1) {
        if (threadIdx.x < s) red[threadIdx.x] += red[threadIdx.x + s];
        __syncthreads();
    }
    if (threadIdx.x == 0) partials[blockIdx.x] = red[0];
}

// ---------------------------------------------------------------------------
// Kernel 4: deterministic final reduction -> commit_loss = mean squared error
// ---------------------------------------------------------------------------
__global__ __launch_bounds__(256) void vq_finalize_kernel(const float* __restrict__ partials,
                                                          float* __restrict__ out_loss) {
    __shared__ float red[256];
    float s = 0.0f;
    for (int i = threadIdx.x; i < 4096; i += 256) s += partials[i];
    red[threadIdx.x] = s;
    __syncthreads();
    #pragma unroll
    for (int st = 128; st > 0; st >>= 1) {
        if (threadIdx.x < st) red[threadIdx.x] += red[threadIdx.x + st];
        __syncthreads();
    }
    if (threadIdx.x == 0) out_loss[0] = red[0] * (1.0f / (float)ZQ_ELEMS);
}

// ---------------------------------------------------------------------------
extern "C" void kernel_launch(void* const* d_in, const int* in_sizes, int n_in,
                              void* d_out, int out_size, void* d_ws, size_t ws_size,
                              hipStream_t stream) {
    const float* h   = (const float*)d_in[0];   // (16, 256, 32, 32) fp32
    const float* emb = (const float*)d_in[1];   // (4096, 256) fp32

    float* out       = (float*)d_out;
    float* out_zq    = out;                       // 4,194,304 floats
    float* out_idx_f = out + ZQ_ELEMS;            // 16,384 floats (indices)
    float* out_loss  = out + ZQ_ELEMS + N_TOTAL;  // 1 float (commit_loss)

    char*  ws       = (char*)d_ws;
    float* partials = (float*)ws;                              // 4096 floats
    int*   indices  = (int*)(ws + 4096 * sizeof(float));       // 16384 ints
    float* enorm    = (float*)(ws + 4096 * sizeof(float)
                                  + N_TOTAL * sizeof(int));    // 4096 floats

    vq_enorm_kernel <<<V_DIM / 8,    256, 0, stream>>>(emb, enorm);
    vq_argmin_kernel<<<N_TOTAL / 32, 256, 0, stream>>>(h, emb, enorm, indices);
    vq_gather_kernel<<<4096,         256, 0, stream>>>(h, emb, indices, out_zq, out_idx_f, partials);
    vq_finalize_kernel<<<1,          256, 0, stream>>>(partials, out_loss);
}